// VectorQuantizer_85684597555500
// MI455X (gfx1250) — compile-verified
//
#include <hip/hip_runtime.h>

typedef float v2f __attribute__((ext_vector_type(2)));
typedef float v8f __attribute__((ext_vector_type(8)));

#define N_EMB   1024
#define EMB_DIM 64
#define NBATCH  32
#define HW      4096               // 64*64
#define N_ROWS  (NBATCH * HW)      // 131072

// d_out layout (floats), reference return order flattened:
#define ZQ_OFF    0
#define LOSS_OFF  8388608
#define PERP_OFF  8388609
#define NEAR_OFF  8388610
#define ZFLAT_OFF (NEAR_OFF + N_ROWS)   // 8519682

// ---------------------------------------------------------------------------
// Kernel 1: [B,C,H,W] -> z_flat [(B*H*W), C].  LDS-tiled transpose.
// Global side is all b128; LDS side scalar b32 with 65-float row pad
// (bank = (c + r) mod 64 -> conflict-free row writes and column reads).
// ---------------------------------------------------------------------------
__global__ void vq_transpose(const float* __restrict__ z,
                             float* __restrict__ zflat) {
    __shared__ float tile[64][65];
    const int b   = blockIdx.x >> 6;          // 0..31
    const int hw0 = (blockIdx.x & 63) << 6;   // 0..4032 step 64
    const int t   = threadIdx.x;

    const float* zb = z + (size_t)b * (EMB_DIM * HW);
    for (int i = t; i < 1024; i += 256) {     // 4096 elems, 4 per thread-iter
        int c = i >> 4, x4 = (i & 15) << 2;
        float4 v = *(const float4*)(zb + c * HW + hw0 + x4);
        tile[c][x4 + 0] = v.x; tile[c][x4 + 1] = v.y;
        tile[c][x4 + 2] = v.z; tile[c][x4 + 3] = v.w;
    }
    __syncthreads();
    float* outb = zflat + (size_t)(b * HW + hw0) * EMB_DIM;
    for (int i = t; i < 1024; i += 256) {
        int r = i >> 4, c4 = (i & 15) << 2;
        float4 v;
        v.x = tile[c4 + 0][r]; v.y = tile[c4 + 1][r];
        v.z = tile[c4 + 2][r]; v.w = tile[c4 + 3][r];
        *(float4*)(outb + r * EMB_DIM + c4) = v;
    }
}

// ---------------------------------------------------------------------------
// Kernel 2: per-code squared norms; zero counts + loss accumulator.
// ---------------------------------------------------------------------------
__global__ void vq_init(const float* __restrict__ emb,
                        float* __restrict__ norms,
                        int* __restrict__ counts,
                        float* __restrict__ lossAcc) {
    int k = blockIdx.x * 256 + threadIdx.x;
    if (k < N_EMB) {
        const float* e = emb + k * EMB_DIM;
        float s = 0.f;
        #pragma unroll
        for (int i = 0; i < EMB_DIM; ++i) s = fmaf(e[i], e[i], s);
        norms[k]  = s;
        counts[k] = 0;
    }
    if (k == 0) *lossAcc = 0.f;
}

// ---------------------------------------------------------------------------
// Kernel 3: WMMA distance + argmin.
//   block = 512 threads (16 waves); each wave owns TWO 16-row A slabs
//   (32 rows) that share every B fetch: halves LDS B-traffic per WMMA and
//   gives two independent accumulation chains (c0/c1) to hide WMMA latency.
//   Full codebook (256KB) + norms (4KB) live in LDS, filled with b128.
//   256 blocks, one per 512 rows.
// ---------------------------------------------------------------------------
__global__ __launch_bounds__(512)
void vq_argmin(const float* __restrict__ zflat,
               const float* __restrict__ emb,
               const float* __restrict__ norms,
               int* __restrict__ counts,
               float* __restrict__ nearest_f) {
    extern __shared__ float smem[];
    float* elds = smem;                       // [1024*64]
    float* nlds = smem + N_EMB * EMB_DIM;     // [1024]

    const int t = threadIdx.x;
    {   // vectorized codebook fill: 16384 float4s
        const float4* s4 = (const float4*)emb;
        float4* d4 = (float4*)elds;
        for (int i = t; i < (N_EMB * EMB_DIM) / 4; i += 512) d4[i] = s4[i];
        for (int i = t; i < N_EMB; i += 512) nlds[i] = norms[i];
    }
    __syncthreads();

    const int wave = t >> 5;        // 0..15
    const int lane = t & 31;
    const int half = lane >> 4;     // K-pair select (and M 0..7 vs 8..15 in C)
    const int lr   = lane & 15;
    const int m0   = blockIdx.x * 512 + wave * 32;

    // A-matrix 16x4 f32 layout (ISA 7.12.2): lanes 0-15 = rows M, v[0]=K0,
    // v[1]=K1; lanes 16-31 same M with K2,K3.  Two slabs preloaded.
    const float* zr0 = zflat + (size_t)(m0 + lr) * EMB_DIM;
    const float* zr1 = zr0 + 16 * EMB_DIM;
    v2f a0[16], a1[16];
    #pragma unroll
    for (int kk = 0; kk < 16; ++kk) {
        a0[kk] = *(const v2f*)(zr0 + 4 * kk + 2 * half);
        a1[kk] = *(const v2f*)(zr1 + 4 * kk + 2 * half);
    }

    float best0[8], best1[8];
    int   bidx0[8], bidx1[8];
    #pragma unroll
    for (int r = 0; r < 8; ++r) {
        best0[r] = 3.4e38f; bidx0[r] = 0;
        best1[r] = 3.4e38f; bidx1[r] = 0;
    }

    for (int j = 0; j < 64; ++j) {            // 64 tiles of 16 codes
        const int n = j * 16 + lr;            // this lane's code column
        const float* erow = elds + n * EMB_DIM;
        const float nrm = nlds[n];
        v8f c0 = {}, c1 = {};
        #pragma unroll
        for (int kk = 0; kk < 16; ++kk) {
            // B-matrix 4x16: B[k][n] = emb[n][k]; same lane-half K pairing.
            v2f bb = *(const v2f*)(erow + 4 * kk + 2 * half);
            c0 = __builtin_amdgcn_wmma_f32_16x16x4_f32(
                     false, a0[kk], false, bb, (short)0, c0, false, false);
            c1 = __builtin_amdgcn_wmma_f32_16x16x4_f32(
                     false, a1[kk], false, bb, (short)0, c1, false, false);
        }
        // C/D layout: lane = N, VGPR r = M=r (lanes 0-15) or M=r+8 (16-31).
        #pragma unroll
        for (int r = 0; r < 8; ++r) {
            float d0 = fmaf(-2.f, c0[r], nrm);
            if (d0 < best0[r]) { best0[r] = d0; bidx0[r] = n; }
            float d1 = fmaf(-2.f, c1[r], nrm);
            if (d1 < best1[r]) { best1[r] = d1; bidx1[r] = n; }
        }
    }

    // Butterfly argmin across the 16 lanes of each half (masks stay < 16).
    #pragma unroll
    for (int m = 8; m >= 1; m >>= 1) {
        #pragma unroll
        for (int r = 0; r < 8; ++r) {
            float od = __shfl_xor(best0[r], m, 32);
            int   oi = __shfl_xor(bidx0[r], m, 32);
            if (od < best0[r] || (od == best0[r] && oi < bidx0[r])) {
                best0[r] = od; bidx0[r] = oi;
            }
            float pd = __shfl_xor(best1[r], m, 32);
            int   pi = __shfl_xor(bidx1[r], m, 32);
            if (pd < best1[r] || (pd == best1[r] && pi < bidx1[r])) {
                best1[r] = pd; bidx1[r] = pi;
            }
        }
    }
    if (lr == 0) {
        #pragma unroll
        for (int r = 0; r < 8; ++r) {
            int row0 = m0 + half * 8 + r;
            nearest_f[row0] = (float)bidx0[r];
            atomicAdd(&counts[bidx0[r]], 1);
            int row1 = m0 + 16 + half * 8 + r;
            nearest_f[row1] = (float)bidx1[r];
            atomicAdd(&counts[bidx1[r]], 1);
        }
    }
}

// ---------------------------------------------------------------------------
// Kernel 4: gather z_q, write back in [B,C,H,W] via LDS tile (b128 global,
// conflict-free b32 LDS), fused squared-error block reduction for the loss.
// ---------------------------------------------------------------------------
__global__ void vq_gather(const float* __restrict__ emb,
                          const float* __restrict__ zflat,
                          const float* __restrict__ nearest_f,
                          float* __restrict__ zq_out,
                          float* __restrict__ lossAcc) {
    __shared__ float tile[64][65];
    __shared__ float red[256];
    const int b   = blockIdx.x >> 6;
    const int hw0 = (blockIdx.x & 63) << 6;
    const int t   = threadIdx.x;
    const int n0  = b * HW + hw0;

    float acc = 0.f;
    for (int i = t; i < 1024; i += 256) {
        int r = i >> 4, c4 = (i & 15) << 2;
        int n = n0 + r;
        int k = (int)nearest_f[n];                 // exact: k < 1024
        float4 q  = *(const float4*)(emb + k * EMB_DIM + c4);
        float4 zv = *(const float4*)(zflat + (size_t)n * EMB_DIM + c4);
        float dx = q.x - zv.x, dy = q.y - zv.y;
        float dz = q.z - zv.z, dw = q.w - zv.w;
        acc = fmaf(dx, dx, acc); acc = fmaf(dy, dy, acc);
        acc = fmaf(dz, dz, acc); acc = fmaf(dw, dw, acc);
        tile[c4 + 0][r] = q.x; tile[c4 + 1][r] = q.y;
        tile[c4 + 2][r] = q.z; tile[c4 + 3][r] = q.w;
    }
    red[t] = acc;
    __syncthreads();

    float* outb = zq_out + (size_t)b * (EMB_DIM * HW) + hw0;
    for (int i = t; i < 1024; i += 256) {
        int c = i >> 4, x4 = (i & 15) << 2;
        float4 v;
        v.x = tile[c][x4 + 0]; v.y = tile[c][x4 + 1];
        v.z = tile[c][x4 + 2]; v.w = tile[c][x4 + 3];
        *(float4*)(outb + c * HW + x4) = v;
    }
    for (int s = 128; s > 0; s >>= 1) {
        if (t < s) red[t] += red[t + s];
        __syncthreads();
    }
    if (t == 0) atomicAdd(lossAcc, red[0]);
}

// ---------------------------------------------------------------------------
// Kernel 5: perplexity from counts + final loss scale.
// ---------------------------------------------------------------------------
__global__ void vq_finalize(const int* __restrict__ counts,
                            const float* __restrict__ lossAcc,
                            float* __restrict__ out) {
    __shared__ float red[1024];
    const int t = threadIdx.x;
    float e = (float)counts[t] * (1.0f / (float)N_ROWS);
    red[t] = e * logf(e + 1e-10f);
    __syncthreads();
    for (int s = 512; s > 0; s >>= 1) {
        if (t < s) red[t] += red[t + s];
        __syncthreads();
    }
    if (t == 0) {
        out[PERP_OFF] = expf(-red[0]);
        out[LOSS_OFF] = 1.25f * (*lossAcc) * (1.0f / 8388608.0f);  // (1+beta)*MSE
    }
}

// ---------------------------------------------------------------------------
extern "C" void kernel_launch(void* const* d_in, const int* in_sizes, int n_in,
                              void* d_out, int out_size, void* d_ws, size_t ws_size,
                              hipStream_t stream) {
    const float* z   = (const float*)d_in[0];  // [32,64,64,64]
    const float* emb = (const float*)d_in[1];  // [1024,64]
    float* out = (float*)d_out;

    float* norms   = (float*)d_ws;             // 1024 f32
    int*   counts  = (int*)d_ws + 1024;        // 1024 i32
    float* lossAcc = (float*)d_ws + 2048;      // 1 f32

    float* zq_out    = out + ZQ_OFF;
    float* nearest_f = out + NEAR_OFF;
    float* zflat     = out + ZFLAT_OFF;

    vq_transpose<<<2048, 256, 0, stream>>>(z, zflat);
    vq_init<<<4, 256, 0, stream>>>(emb, norms, counts, lossAcc);

    size_t lds_bytes = (size_t)(N_EMB * EMB_DIM + N_EMB) * sizeof(float); // 256KB+4KB
    vq_argmin<<<256, 512, lds_bytes, stream>>>(zflat, emb, norms, counts, nearest_f);

    vq_gather<<<2048, 256, 0, stream>>>(emb, zflat, nearest_f, zq_out, lossAcc);
    vq_finalize<<<1, 1024, 0, stream>>>(counts, lossAcc, out);
}